// EOSGenerator_3504693314120
// MI455X (gfx1250) — compile-verified
//
#include <hip/hip_runtime.h>
#include <hip/hip_bf16.h>
#include <math.h>

#define EMB   128
#define HID   512
#define BATCH 8192
#define WIN   128
#define CI    (4 * HID)       // 2048
#define WLIN  (CI + EMB)      // 2176

typedef __attribute__((ext_vector_type(2))) float v2f;
typedef __attribute__((ext_vector_type(4))) float v4f;
typedef __attribute__((ext_vector_type(8))) float v8f;

// ---------------------------------------------------------------------------
// Kernel 1: T[b,f] = sum_e ctr[b,e] * Wb[e,f]   (8192x128 = 8192x128 @ 128x128)
// fp32 WMMA 16x16x4. 8 waves/block, each wave -> one 16x16 tile (full N=128).
// ---------------------------------------------------------------------------
__global__ __launch_bounds__(256) void eos_gemm_T(const float* __restrict__ ctr,
                                                  const float* __restrict__ Wb,
                                                  float* __restrict__ T) {
  const int lane = threadIdx.x & 31;
  const int wave = threadIdx.x >> 5;
  const int m0   = blockIdx.x * 16;   // 512 blocks cover 8192 rows
  const int n0   = wave * 16;         // 8 waves cover 128 cols
  const int half = lane >> 4;         // 0: K pair {0,1}; 1: K pair {2,3}
  const int l    = lane & 15;

  v8f acc = {};
  // A 16x4 f32: lanes0-15 -> v0=K0, v1=K1 ; lanes16-31 -> v0=K2, v1=K3 (M = lane&15)
  // B 4x16 f32: lanes0-15 -> v0=K0 row, v1=K1 row ; lanes16-31 -> K2/K3 (N = lane&15)
  const int m = m0 + l;
  const int n = n0 + l;
  for (int k0 = 0; k0 < EMB; k0 += 4) {
    v2f a, b;
    a.x = ctr[(size_t)m * EMB + k0 + half * 2 + 0];
    a.y = ctr[(size_t)m * EMB + k0 + half * 2 + 1];
    b.x = Wb[(size_t)(k0 + half * 2 + 0) * EMB + n];
    b.y = Wb[(size_t)(k0 + half * 2 + 1) * EMB + n];
    acc = __builtin_amdgcn_wmma_f32_16x16x4_f32(
        /*neg_a=*/false, a, /*neg_b=*/false, b,
        /*c_mod=*/(short)0, acc, /*reuse_a=*/false, /*reuse_b=*/false);
  }
  // C/D 16x16 f32: VGPR r -> (M=r, N=lane) lanes0-15 ; (M=r+8, N=lane-16) lanes16-31
#pragma unroll
  for (int r = 0; r < 8; ++r) {
    T[(size_t)(m0 + r + half * 8) * EMB + n] = acc[r];
  }
}

// ---------------------------------------------------------------------------
// Kernel 2: per-row fused masked-softmax attention + final linear.
// One block (256 thr = 8 wave32) per batch row; window[b] staged in LDS once.
// ---------------------------------------------------------------------------
__global__ __launch_bounds__(256) void eos_fused(const float* __restrict__ context_info,
                                                 const float* __restrict__ window,
                                                 const int*   __restrict__ window_mask,
                                                 const float* __restrict__ T,
                                                 const float* __restrict__ W_linear,
                                                 const float* __restrict__ b_linear,
                                                 float* __restrict__ out) {
  __shared__ float win[WIN * EMB];   // 64 KB: whole window[b]
  __shared__ float t_lds[EMB];
  __shared__ float e_lds[WIN];
  __shared__ float red[256];
  __shared__ float s_max, s_inv;

  const int    tid  = threadIdx.x;
  const int    lane = tid & 31;
  const int    wave = tid >> 5;
  const size_t b    = blockIdx.x;

  // ---- stage window[b] into LDS (non-temporal: stream 512MB through once) ----
  {
    const v4f* src = (const v4f*)(window + b * (size_t)(WIN * EMB));
    v4f*       dst = (v4f*)win;
#pragma unroll
    for (int i = 0; i < (WIN * EMB / 4) / 256; ++i) {   // 16 x 128-bit per thread
      const int idx = tid + i * 256;
      dst[idx] = __builtin_nontemporal_load(&src[idx]);
    }
  }
  if (tid < EMB) t_lds[tid] = T[b * EMB + tid];
  __syncthreads();

  // ---- pass 1: e[w] = dot(t, window[b,w,:]), mask -> -inf ----
  for (int i = 0; i < WIN / 8; ++i) {
    const int w   = wave * (WIN / 8) + i;
    float     acc = 0.f;
#pragma unroll
    for (int j = 0; j < EMB / 32; ++j) {
      const int f = lane + j * 32;
      acc += t_lds[f] * win[w * EMB + f];
    }
#pragma unroll
    for (int off = 16; off > 0; off >>= 1) acc += __shfl_xor(acc, off, 32);
    if (lane == 0) {
      const int mv = window_mask[b * WIN + w];
      e_lds[w]     = mv ? acc : -INFINITY;
    }
  }
  __syncthreads();

  // ---- masked softmax over 128 slots ----
  if (wave == 0) {
    float m = fmaxf(fmaxf(e_lds[lane], e_lds[lane + 32]),
                    fmaxf(e_lds[lane + 64], e_lds[lane + 96]));
#pragma unroll
    for (int off = 16; off > 0; off >>= 1) m = fmaxf(m, __shfl_xor(m, off, 32));
    if (lane == 0) s_max = m;
  }
  __syncthreads();
  if (tid < WIN) e_lds[tid] = __expf(e_lds[tid] - s_max);  // exp(-inf)=0 for masked
  __syncthreads();
  if (wave == 0) {
    float s = e_lds[lane] + e_lds[lane + 32] + e_lds[lane + 64] + e_lds[lane + 96];
#pragma unroll
    for (int off = 16; off > 0; off >>= 1) s += __shfl_xor(s, off, 32);
    if (lane == 0) s_inv = 1.f / s;
  }
  __syncthreads();
  if (tid < WIN) e_lds[tid] *= s_inv;
  __syncthreads();

  // ---- pass 2: context_eos[e] = sum_w at[w]*win[w,e] (split w over 2 halves) ----
  {
    const int e  = tid & (EMB - 1);
    const int wh = tid >> 7;  // 0 or 1
    float     acc = 0.f;
#pragma unroll
    for (int j = 0; j < WIN / 2; ++j) {
      const int w = wh * (WIN / 2) + j;
      acc += e_lds[w] * win[w * EMB + e];
    }
    red[tid] = acc;
  }
  __syncthreads();

  float part = 0.f;
  if (tid < EMB) {
    const float ceos = red[tid] + red[tid + 128];
    part = ceos * W_linear[CI + tid];
  }
  __syncthreads();  // done reading red before reuse

  // ---- context_info dot (non-temporal stream, coalesced float4) ----
  {
    const v4f* ci = (const v4f*)(context_info + b * (size_t)CI);
    const v4f* wl = (const v4f*)W_linear;
#pragma unroll
    for (int j = 0; j < (CI / 4) / 256; ++j) {  // 2 x float4 per thread
      const int idx = tid + j * 256;
      const v4f a  = __builtin_nontemporal_load(&ci[idx]);
      const v4f w4 = wl[idx];
      part += a.x * w4.x + a.y * w4.y + a.z * w4.z + a.w * w4.w;
    }
  }
  red[tid] = part;
  __syncthreads();
#pragma unroll
  for (int s = 128; s > 0; s >>= 1) {
    if (tid < s) red[tid] += red[tid + s];
    __syncthreads();
  }
  if (tid == 0) out[b] = red[0] + b_linear[0];
}

// ---------------------------------------------------------------------------
extern "C" void kernel_launch(void* const* d_in, const int* in_sizes, int n_in,
                              void* d_out, int out_size, void* d_ws, size_t ws_size,
                              hipStream_t stream) {
  const float* context_info = (const float*)d_in[0];  // [B, 2048]
  const float* window       = (const float*)d_in[1];  // [B, 128, 128]
  const int*   window_mask  = (const int*)d_in[2];    // [B, 128] (bool -> int)
  const float* ctr          = (const float*)d_in[3];  // [B, 128]
  const float* W_bilinear   = (const float*)d_in[4];  // [128, 128]
  const float* W_linear     = (const float*)d_in[5];  // [1, 2176]
  const float* b_linear     = (const float*)d_in[6];  // [1]
  float*       out          = (float*)d_out;          // [B, 1]
  float*       T            = (float*)d_ws;           // [B, 128] = 4 MB scratch

  eos_gemm_T<<<BATCH / 16, 256, 0, stream>>>(ctr, W_bilinear, T);
  eos_fused<<<BATCH, 256, 0, stream>>>(context_info, window, window_mask, T,
                                       W_linear, b_linear, out);
}